// DigitCaps_82678120448503
// MI455X (gfx1250) — compile-verified
//
#include <hip/hip_runtime.h>

// ---------------- problem constants ----------------
#define IN_CAPS   1152
#define IN_SIZE   8
#define OUT_CAPS  10
#define OUT_SIZE  16
#define NOUT      (OUT_CAPS * OUT_SIZE)   // 160
#define BATCH     1024
#define EPSQ      1e-8f

// ---------------- tiling ----------------
#define BT     16                 // batch tile (WMMA N dim)
#define NBT    (BATCH / BT)       // 64 batch tiles
#define ICH    8                  // i-chunks (partial-s slots)
#define IPC    (IN_CAPS / ICH)    // 144 input caps per chunk
#define WAVES  8                  // waves per workgroup (256 threads)
#define IPW    (IPC / WAVES)      // 18 input caps per wave

typedef __attribute__((ext_vector_type(2))) float v2f;
typedef __attribute__((ext_vector_type(8))) float v8f;

// D(16x16,f32) = A(16x4,f32) x B(4x16,f32) + C   — full-precision f32 WMMA
__device__ __forceinline__ v8f wmma_f32_4(v2f a, v2f b, v8f c) {
  return __builtin_amdgcn_wmma_f32_16x16x4_f32(false, a, false, b, (short)0, c,
                                               false, false);
}

// One routing sweep: recompute u_hat^T tiles via WMMA, form logits
// b0[i,j] + <u_hat[b,i,j,:], vsum[b,j,:]>, softmax over j, accumulate
// s[b,j,d] partials per i-chunk.
//
// WMMA orientation per input capsule i and output tile j:
//   D[M,N] : M = d (output dim within capsule j), N = b (batch in tile)
//   A[M,K] = W[i, s, j*16+M]   (K = s)
//   B[K,N] = x[b, i, s]
// Lane mapping (wave32): b = lane&15, half h = lane>>4.
//   A frag reg v : K = v + 2h   | D frag reg r : M = r + 8h, N = lane&15
__global__ __launch_bounds__(256) void caps_route_sweep(
    const float* __restrict__ x,      // [B][1152][8]
    const float* __restrict__ Wm,     // [1152][8][160]
    const float* __restrict__ bij,    // [1152][10]
    const float* __restrict__ vsum,   // [B][10][16]  (sum of previous v's)
    float* __restrict__ s_part,       // [ICH][B][10][16]
    int use_vsum) {
  __shared__ float vsum_s[NOUT * BT];   // [j*16+d][b]
  __shared__ float s_s[NOUT * BT];      // [j*16+d][b]

  const int tid  = threadIdx.x;
  const int lane = tid & 31;
  const int wave = tid >> 5;
  const int ic   = blockIdx.x;          // i-chunk
  const int bt   = blockIdx.y;          // batch tile
  const int b    = lane & 15;
  const int h    = lane >> 4;

  // Stage vsum for this batch tile into LDS ([jd][b] layout), zero s accum.
  for (int idx = tid; idx < NOUT * BT; idx += 256) {
    const int bb = idx / NOUT;
    const int jd = idx % NOUT;
    vsum_s[jd * BT + bb] =
        use_vsum ? vsum[(size_t)(bt * BT + bb) * NOUT + jd] : 0.0f;
    s_s[idx] = 0.0f;
  }
  __syncthreads();

  v8f sacc[OUT_CAPS];
#pragma unroll
  for (int j = 0; j < OUT_CAPS; ++j) sacc[j] = (v8f){0, 0, 0, 0, 0, 0, 0, 0};

  for (int k = 0; k < IPW; ++k) {
    const int i = ic * IPC + wave * IPW + k;

    // B fragments from x: reg v -> s = kbase + v + 2h  (contiguous pair)
    const float* xp =
        x + ((size_t)(bt * BT + b) * IN_CAPS + i) * IN_SIZE + 2 * h;
    const v2f xb0 = *(const v2f*)(xp);       // s = 2h, 2h+1
    const v2f xb1 = *(const v2f*)(xp + 4);   // s = 2h+4, 2h+5

    // u_hat^T fragments, one 16x16 tile per output capsule j
    v8f uf[OUT_CAPS];
    const float* wp = Wm + (size_t)i * (IN_SIZE * NOUT) + (lane & 15);
#pragma unroll
    for (int j = 0; j < OUT_CAPS; ++j) {
      const float* wq = wp + j * OUT_SIZE;
      const v2f a0 = {wq[(2 * h + 0) * NOUT], wq[(2 * h + 1) * NOUT]};
      const v2f a1 = {wq[(2 * h + 4) * NOUT], wq[(2 * h + 5) * NOUT]};
      v8f acc = (v8f){0, 0, 0, 0, 0, 0, 0, 0};
      acc = wmma_f32_4(a0, xb0, acc);   // K = 0..3
      acc = wmma_f32_4(a1, xb1, acc);   // K = 4..7
      uf[j] = acc;
    }

    // Routing logits: b0[i,j] + sum_d u_hat[b,i,j,d] * vsum[b,j,d]
    float logit[OUT_CAPS];
#pragma unroll
    for (int j = 0; j < OUT_CAPS; ++j) {
      float t = 0.0f;
      if (use_vsum) {
#pragma unroll
        for (int r = 0; r < 8; ++r)
          t = __builtin_fmaf(uf[j][r], vsum_s[(j * 16 + r + 8 * h) * BT + b],
                             t);
        t += __shfl_xor(t, 16, 32);  // combine d-halves
      }
      logit[j] = t + bij[i * OUT_CAPS + j];
    }

    // softmax over j (per-lane scalar work, lane = batch element)
    float m = logit[0];
#pragma unroll
    for (int j = 1; j < OUT_CAPS; ++j) m = fmaxf(m, logit[j]);
    float e[OUT_CAPS];
    float esum = 0.0f;
#pragma unroll
    for (int j = 0; j < OUT_CAPS; ++j) {
      e[j] = __expf(logit[j] - m);
      esum += e[j];
    }
    const float inv = 1.0f / esum;

    // s[b,j,d] += c[b,j] * u_hat[b,j,d]  (c is a per-lane scalar)
#pragma unroll
    for (int j = 0; j < OUT_CAPS; ++j) {
      const float cj = e[j] * inv;
#pragma unroll
      for (int r = 0; r < 8; ++r)
        sacc[j][r] = __builtin_fmaf(cj, uf[j][r], sacc[j][r]);
    }
  }

  // Deterministic cross-wave reduction into LDS (sequential wave order).
  for (int w = 0; w < WAVES; ++w) {
    if (wave == w) {
#pragma unroll
      for (int j = 0; j < OUT_CAPS; ++j)
#pragma unroll
        for (int r = 0; r < 8; ++r)
          s_s[(j * 16 + r + 8 * h) * BT + b] += sacc[j][r];
    }
    __syncthreads();
  }

  // Emit this chunk's partial s (coalesced, non-atomic: exclusive slot).
  for (int idx = tid; idx < NOUT * BT; idx += 256) {
    const int bb = idx / NOUT;
    const int jd = idx % NOUT;
    s_part[((size_t)ic * BATCH + bt * BT + bb) * NOUT + jd] =
        s_s[jd * BT + bb];
  }
}

// Sum chunk partials, squash, update vsum, emit final v on last sweep.
__global__ __launch_bounds__(256) void caps_squash(
    const float* __restrict__ s_part,  // [ICH][B][10][16]
    float* __restrict__ vsum,          // [B][10][16]
    float* __restrict__ out,           // [B][10][16]
    int t) {
  const int idx = blockIdx.x * blockDim.x + threadIdx.x;  // (b*10 + j)
  if (idx >= BATCH * OUT_CAPS) return;
  const int bg = idx / OUT_CAPS;
  const int j  = idx % OUT_CAPS;

  float s[OUT_SIZE];
#pragma unroll
  for (int d = 0; d < OUT_SIZE; ++d) s[d] = 0.0f;
  for (int k = 0; k < ICH; ++k) {
    const float* p = s_part + ((size_t)k * BATCH + bg) * NOUT + j * OUT_SIZE;
#pragma unroll
    for (int d = 0; d < OUT_SIZE; ++d) s[d] += p[d];
  }

  float sq = 0.0f;
#pragma unroll
  for (int d = 0; d < OUT_SIZE; ++d) sq += s[d] * s[d];
  const float scale = sq / ((1.0f + sq) * sqrtf(sq + EPSQ));

  if (t == 3) {
    float* o = out + (size_t)bg * NOUT + j * OUT_SIZE;
#pragma unroll
    for (int d = 0; d < OUT_SIZE; ++d) o[d] = s[d] * scale;
  } else {
    float* vp = vsum + (size_t)bg * NOUT + j * OUT_SIZE;
    if (t == 0) {
#pragma unroll
      for (int d = 0; d < OUT_SIZE; ++d) vp[d] = s[d] * scale;
    } else {
#pragma unroll
      for (int d = 0; d < OUT_SIZE; ++d) vp[d] += s[d] * scale;
    }
  }
}

extern "C" void kernel_launch(void* const* d_in, const int* in_sizes, int n_in,
                              void* d_out, int out_size, void* d_ws,
                              size_t ws_size, hipStream_t stream) {
  const float* x   = (const float*)d_in[0];  // (1024,1152,8) f32
  const float* Wm  = (const float*)d_in[1];  // (1152,8,160)  f32
  const float* bij = (const float*)d_in[2];  // (1152,10)     f32
  float* out = (float*)d_out;                // (1024,10,16)  f32

  // workspace: vsum (B*160) + s_part (ICH*B*160)  ~= 5.9 MB
  float* vsum   = (float*)d_ws;
  float* s_part = vsum + (size_t)BATCH * NOUT;

  const dim3 gs(ICH, NBT), bs(256);
  const dim3 gq((BATCH * OUT_CAPS + 255) / 256), bq(256);

  for (int t = 0; t < 4; ++t) {
    caps_route_sweep<<<gs, bs, 0, stream>>>(x, Wm, bij, vsum, s_part,
                                            t > 0 ? 1 : 0);
    caps_squash<<<gq, bq, 0, stream>>>(s_part, vsum, out, t);
  }
}